// PointCloudLoss_49933289783515
// MI455X (gfx1250) — compile-verified
//
#include <hip/hip_runtime.h>
#include <math.h>

// Problem constants from the reference: B=8, N=4096, D=2.
#define B_SZ 8
#define N_PTS 4096
#define NT (N_PTS / 16)        // 256 16-wide tiles per cloud
#define WG_PER_CLOUD 8         // workgroups per (batch, direction)
#define WAVES 8                // waves per workgroup (wave32)
#define THREADS (WAVES * 32)

typedef __attribute__((ext_vector_type(2))) float v2f;
typedef __attribute__((ext_vector_type(8))) float v8f;

__global__ void zero_out_kernel(float* out) { out[0] = 0.0f; }

// One-directional Chamfer: for each query point q in Q, min over all targets t in T
// of ||q - t||^2, summed over queries, scaled, atomically added to *out.
//
// Distance tile via a single V_WMMA_F32_16X16X4_F32:
//   a_q = (x_q, y_q, ||q||^2, 1),  b_t = (-2x_t, -2y_t, 1, ||t||^2)
//   a_q . b_t = ||q - t||^2
__global__ __launch_bounds__(THREADS)
void chamfer_dir_kernel(const float* __restrict__ Qraw,
                        const float* __restrict__ Traw,
                        float* __restrict__ out, float scale)
{
    // B-operand table in exact WMMA lane layout: [tile j][lane][2 floats] = 64 KB
    __shared__ float bop[NT * 32 * 2];

    const int b  = blockIdx.x / WG_PER_CLOUD;
    const int wg = blockIdx.x % WG_PER_CLOUD;
    const float2* Q = (const float2*)Qraw + (size_t)b * N_PTS;
    const float2* T = (const float2*)Traw + (size_t)b * N_PTS;
    const int tid  = threadIdx.x;
    const int lane = tid & 31;
    const int wave = tid >> 5;

    // ---- Build B operands in LDS ----
    // B 4x16 (KxN) 32-bit layout (mirror of A 16x4):
    //   VGPR0: lanes 0-15 -> K=0 (=-2x), lanes 16-31 -> K=2 (=1)
    //   VGPR1: lanes 0-15 -> K=1 (=-2y), lanes 16-31 -> K=3 (=||t||^2)
    for (int idx = tid; idx < NT * 32; idx += THREADS) {
        const int j = idx >> 5;
        const int l = idx & 31;
        const float2 tm = T[j * 16 + (l & 15)];
        float e0, e1;
        if (l < 16) { e0 = -2.0f * tm.x; e1 = -2.0f * tm.y; }
        else        { e0 = 1.0f;         e1 = tm.x * tm.x + tm.y * tm.y; }
        bop[idx * 2 + 0] = e0;
        bop[idx * 2 + 1] = e1;
    }
    __syncthreads();

    const int tilesPerWG = NT / WG_PER_CLOUD; // 32 query tiles per WG
    float waveAcc = 0.0f;

    for (int qi = wg * tilesPerWG + wave; qi < (wg + 1) * tilesPerWG; qi += WAVES) {
        // ---- A operand: 16 query rows, K-padded with (||q||^2, 1) ----
        const float2 qv = Q[qi * 16 + (lane & 15)];
        v2f a;
        if (lane < 16) { a[0] = qv.x; a[1] = qv.y; }
        else           { a[0] = qv.x * qv.x + qv.y * qv.y; a[1] = 1.0f; }

        // Two independent accumulator sets for ILP across the WMMA->VALU hazard.
        v8f acc0, acc1;
#pragma unroll
        for (int r = 0; r < 8; ++r) { acc0[r] = 3.0e38f; acc1[r] = 3.0e38f; }

        for (int j = 0; j < NT; j += 2) {
            const v2f b0 = *(const v2f*)&bop[((j + 0) * 32 + lane) * 2];
            const v2f b1 = *(const v2f*)&bop[((j + 1) * 32 + lane) * 2];
            v8f c0 = {};
            v8f c1 = {};
            // (neg_a, A, neg_b, B, c_mod, C, reuse_a, reuse_b)
            const v8f d0 = __builtin_amdgcn_wmma_f32_16x16x4_f32(
                false, a, false, b0, (short)0, c0, false, false);
            const v8f d1 = __builtin_amdgcn_wmma_f32_16x16x4_f32(
                false, a, false, b1, (short)0, c1, false, false);
#pragma unroll
            for (int r = 0; r < 8; ++r) {
                acc0[r] = fminf(acc0[r], d0[r]);
                acc1[r] = fminf(acc1[r], d1[r]);
            }
        }

        // D layout: lane<16 holds rows r=0..7 (col = lane); lane>=16 holds rows
        // r+8 (col = lane-16). xor-reduce with masks 1/2/4/8 stays inside each
        // 16-lane half -> per-row min. Clamp at 0 once (min/max(.,0) commute).
        float rowsum = 0.0f;
#pragma unroll
        for (int r = 0; r < 8; ++r) {
            float m = fminf(acc0[r], acc1[r]);
            m = fminf(m, __shfl_xor(m, 1, 32));
            m = fminf(m, __shfl_xor(m, 2, 32));
            m = fminf(m, __shfl_xor(m, 4, 32));
            m = fminf(m, __shfl_xor(m, 8, 32));
            rowsum += fmaxf(m, 0.0f);
        }
        // lanes 0-15: sum(rows 0..7); lanes 16-31: sum(rows 8..15)
        waveAcc += __shfl(rowsum, 0, 32) + __shfl(rowsum, 16, 32);
    }

    if (lane == 0) {
        unsafeAtomicAdd(out, waveAcc * scale);
    }
}

// EMD approximation term: mean over all B*N*D of (cumsum(true)-cumsum(pred))^2.
// For D=2: c0 = tx - px, c1 = (tx+ty) - (px+py).
__global__ void emd_kernel(const float* __restrict__ traw,
                           const float* __restrict__ praw,
                           float* __restrict__ out)
{
    const int i = blockIdx.x * blockDim.x + threadIdx.x; // point index over B*N
    float v = 0.0f;
    if (i < B_SZ * N_PTS) {
        const float2 tv = ((const float2*)traw)[i];
        const float2 pv = ((const float2*)praw)[i];
        const float c0 = tv.x - pv.x;
        const float c1 = (tv.x + tv.y) - (pv.x + pv.y);
        v = c0 * c0 + c1 * c1;
    }
#pragma unroll
    for (int off = 16; off > 0; off >>= 1) v += __shfl_xor(v, off, 32);
    if ((threadIdx.x & 31) == 0) {
        unsafeAtomicAdd(out, v * (1.0f / (float)(B_SZ * N_PTS * 2)));
    }
}

extern "C" void kernel_launch(void* const* d_in, const int* in_sizes, int n_in,
                              void* d_out, int out_size, void* d_ws, size_t ws_size,
                              hipStream_t stream)
{
    const float* y_true = (const float*)d_in[0];
    const float* y_pred = (const float*)d_in[1];
    float* out = (float*)d_out;

    zero_out_kernel<<<1, 1, 0, stream>>>(out);

    const dim3 grid(B_SZ * WG_PER_CLOUD);
    const dim3 block(THREADS);
    // forward: queries = y_true, targets = y_pred; cd = mean over batch of row-min sums
    chamfer_dir_kernel<<<grid, block, 0, stream>>>(y_true, y_pred, out, 1.0f / (float)B_SZ);
    // backward: swap roles
    chamfer_dir_kernel<<<grid, block, 0, stream>>>(y_pred, y_true, out, 1.0f / (float)B_SZ);

    emd_kernel<<<(B_SZ * N_PTS + 255) / 256, 256, 0, stream>>>(y_true, y_pred, out);
}